// CoordAtt3D_24550033064654
// MI455X (gfx1250) — compile-verified
//
#include <hip/hip_runtime.h>
#include <hip/hip_bf16.h>

// CoordAtt3D for MI455X (gfx1250, wave32).
// N=4, C=64, D=H=W=64, MIP=8. x is 256MB f32 -> pure HBM-bound problem.
// Roofline: 256MB (pool read) + 256MB (apply read, partially L2-hit via
// reverse traversal) + 256MB (NT store) ~= 33us @ 23.3 TB/s.
// The middle two einsums are tiny; done with V_WMMA_F32_16X16X4_F32 to keep
// f32 precision while exercising the CDNA5 matrix pipe.

typedef float v2f __attribute__((ext_vector_type(2)));
typedef float v4f __attribute__((ext_vector_type(4)));
typedef float v8f __attribute__((ext_vector_type(8)));

#define NB    4
#define CB    64
#define DIM   64
#define LTOT  192           // 3*DIM pooled length
#define VOL   (DIM*DIM*DIM) // 262144 elements per (n,c)

// ---------------------------------------------------------------------------
// Kernel 1: triple directional mean-pool.
// One block per (n,c) volume (1MB). 256 threads = 8 waves.
// Thread t: w4 = t&15 (float4 column along w), h0 = t>>4 (h = h0+16j).
// xd[d]: per-d wave reduce + ds_add_f32; xh/xw: per-thread partials + LDS atomics.
// Default (RT) loads on purpose: leave the tail of x hot in the 192MB L2 for
// the reverse-order apply pass.
// ---------------------------------------------------------------------------
__global__ void __launch_bounds__(256) pool_kernel(const float* __restrict__ x,
                                                   float* __restrict__ pool) {
  __shared__ float sd[DIM];
  __shared__ float sh[DIM];
  __shared__ float sw[DIM];

  const int nc = blockIdx.x;          // 0..255
  const int t  = threadIdx.x;         // 0..255
  const int w4 = t & 15;              // which float4 along w
  const int h0 = t >> 4;              // base h row
  const int lane = t & 31;

  if (t < DIM) { sd[t] = 0.0f; sh[t] = 0.0f; sw[t] = 0.0f; }
  __syncthreads();

  const v4f* __restrict__ xv = (const v4f*)(x + (size_t)nc * VOL);

  v4f  accw = {0.0f, 0.0f, 0.0f, 0.0f};
  float acch[4] = {0.0f, 0.0f, 0.0f, 0.0f};

  for (int d = 0; d < DIM; ++d) {
    float pd = 0.0f;
    #pragma unroll
    for (int j = 0; j < 4; ++j) {
      const int h = h0 + 16 * j;
      v4f v = xv[(size_t)(d * DIM + h) * 16 + w4];   // global_load_b128
      accw += v;
      float s = (v.x + v.y) + (v.z + v.w);
      acch[j] += s;
      pd += s;
    }
    // wave32 tree reduction for this d-slice
    #pragma unroll
    for (int off = 16; off > 0; off >>= 1) pd += __shfl_down(pd, off, 32);
    if (lane == 0) atomicAdd(&sd[d], pd);            // ds_add_f32
  }

  #pragma unroll
  for (int j = 0; j < 4; ++j) atomicAdd(&sh[h0 + 16 * j], acch[j]);
  atomicAdd(&sw[w4 * 4 + 0], accw.x);
  atomicAdd(&sw[w4 * 4 + 1], accw.y);
  atomicAdd(&sw[w4 * 4 + 2], accw.z);
  atomicAdd(&sw[w4 * 4 + 3], accw.w);
  __syncthreads();

  if (t < LTOT) {
    float v = (t < 64) ? sd[t] : ((t < 128) ? sh[t - 64] : sw[t - 128]);
    pool[(size_t)nc * LTOT + t] = v * (1.0f / 4096.0f);
  }
}

// ---------------------------------------------------------------------------
// Kernel 2: fused middle on the WMMA f32 pipe. 48 blocks x 32 threads (1 wave).
// Block = (n, l_tile of 16). Stage 1: y[m=0..7][l] = sum_c pool[n][c][l]*w1[m][c]
// as 16 x V_WMMA_F32_16X16X4_F32 (K=64 in steps of 4, M padded 8->16).
// Epilogue bias+BN+hardswish in registers. Stage 2: per o-tile (4x16=64 outs),
// K=m=8 in 2 WMMA steps; B operand repacked from the D layout via __shfl.
// VGPR layouts per ISA 7.12.2:
//   A 16x4 f32 : lane holds A[M=lane%16][K=2*(lane/16)+v], v in {0,1}
//   B 4x16 f32 : lane holds B[K=2*(lane/16)+v][N=lane%16]
//   C/D 16x16  : lane holds D[M=v+8*(lane/16)][N=lane%16], v in 0..7
// ---------------------------------------------------------------------------
__global__ void __launch_bounds__(32) mid_kernel(
    const float* __restrict__ pool,
    const float* __restrict__ w1,  const float* __restrict__ b1,
    const float* __restrict__ gma, const float* __restrict__ bta,
    const float* __restrict__ mu,  const float* __restrict__ var,
    const float* __restrict__ wd,  const float* __restrict__ bd,
    const float* __restrict__ wh,  const float* __restrict__ bh,
    const float* __restrict__ ww,  const float* __restrict__ bw,
    float* __restrict__ att) {
  const int blk    = blockIdx.x;     // 0..47
  const int n      = blk / 12;
  const int lt     = blk % 12;
  const int l_base = lt * 16;
  const int axis   = lt >> 2;        // 0:d 1:h 2:w  (uniform per block)
  const float* __restrict__ w2 = (axis == 0) ? wd : ((axis == 1) ? wh : ww);
  const float* __restrict__ b2 = (axis == 0) ? bd : ((axis == 1) ? bh : bw);

  const int lane = threadIdx.x;      // 0..31
  const int lo16 = lane & 15;
  const int hi   = lane >> 4;        // 0 or 1

  // ---- stage 1: conv1 via WMMA f32 16x16x4, accumulate over K=c ----
  const float mmask = (lo16 < 8) ? 1.0f : 0.0f;   // pad M rows 8..15 with zeros
  const int   mrow  = lo16 & 7;
  const float* __restrict__ pn = pool + (size_t)n * CB * LTOT + l_base + lo16;

  v8f acc = {0.0f, 0.0f, 0.0f, 0.0f, 0.0f, 0.0f, 0.0f, 0.0f};
  #pragma unroll
  for (int k = 0; k < CB; k += 4) {
    const int ka = k + 2 * hi;
    v2f A; A.x = w1[mrow * CB + ka] * mmask;  A.y = w1[mrow * CB + ka + 1] * mmask;
    v2f B; B.x = pn[(size_t)ka * LTOT];       B.y = pn[(size_t)(ka + 1) * LTOT];
    acc = __builtin_amdgcn_wmma_f32_16x16x4_f32(
        /*neg_a=*/false, A, /*neg_b=*/false, B, (short)0, acc,
        /*reuse_a=*/false, /*reuse_b=*/false);
  }

  // ---- epilogue: +bias, BN, hardswish; lanes 0-15 hold m=v (valid), hi lanes pad ----
  float y[8];
  #pragma unroll
  for (int v = 0; v < 8; ++v) {
    const int m = v;                                  // valid when hi==0
    float sc = gma[m] * rsqrtf(var[m] + 1e-5f);
    float yv = (acc[v] + b1[m] - mu[m]) * sc + bta[m];
    float cl = fminf(fmaxf(yv + 3.0f, 0.0f), 6.0f);   // hardswish
    yv = yv * cl * (1.0f / 6.0f);
    y[v] = (hi == 0) ? yv : 0.0f;
  }

  // ---- stage 2: a[o][l] = sigmoid(sum_m y[m][l]*w2[o][m] + b2[o]) ----
  #pragma unroll
  for (int ot = 0; ot < 4; ++ot) {
    v8f acc2 = {0.0f, 0.0f, 0.0f, 0.0f, 0.0f, 0.0f, 0.0f, 0.0f};
    #pragma unroll
    for (int ks = 0; ks < 2; ++ks) {
      const int mb = ks * 4;
      // Repack y (D layout, m in vgpr index of lanes 0-15) into B layout:
      // lane needs y[m = 2*hi + v + mb][N = lane%16], source lane = lane%16.
      float t0lo = __shfl(y[mb + 0], lo16, 32);
      float t0hi = __shfl(y[mb + 2], lo16, 32);
      float t1lo = __shfl(y[mb + 1], lo16, 32);
      float t1hi = __shfl(y[mb + 3], lo16, 32);
      v2f B2; B2.x = hi ? t0hi : t0lo;  B2.y = hi ? t1hi : t1lo;
      const int o  = ot * 16 + lo16;    // A: M = o
      const int ma = mb + 2 * hi;       // A: K = m
      v2f A2; A2.x = w2[o * 8 + ma];  A2.y = w2[o * 8 + ma + 1];
      acc2 = __builtin_amdgcn_wmma_f32_16x16x4_f32(
          false, A2, false, B2, (short)0, acc2, false, false);
    }
    #pragma unroll
    for (int v = 0; v < 8; ++v) {
      const int o = ot * 16 + v + 8 * hi;
      float r = acc2[v] + b2[o];
      r = 1.0f / (1.0f + __expf(-r));                 // sigmoid
      att[((size_t)n * CB + o) * LTOT + l_base + lo16] = r;
    }
  }
}

// ---------------------------------------------------------------------------
// Kernel 3: out = x * ad[d] * ah[h] * aw[w]. One block per (n,c), traversed in
// REVERSE (blocks and d/h loops) so reads hit the x-tail left in L2 by kernel 1.
// b128 loads, nontemporal b128 stores (output has no reuse).
// ---------------------------------------------------------------------------
__global__ void __launch_bounds__(256) apply_kernel(const float* __restrict__ x,
                                                    const float* __restrict__ att,
                                                    float* __restrict__ out) {
  const int nc = (NB * CB - 1) - blockIdx.x;   // reverse order
  __shared__ float s_ad[DIM];
  __shared__ float s_ah[DIM];
  __shared__ float s_aw[DIM];

  const int t = threadIdx.x;
  const float* __restrict__ a = att + (size_t)nc * LTOT;
  if (t < DIM) { s_ad[t] = a[t]; s_ah[t] = a[DIM + t]; s_aw[t] = a[2 * DIM + t]; }
  __syncthreads();

  const int w4 = t & 15;
  const int h0 = t >> 4;
  v4f aw4;
  aw4.x = s_aw[w4 * 4 + 0]; aw4.y = s_aw[w4 * 4 + 1];
  aw4.z = s_aw[w4 * 4 + 2]; aw4.w = s_aw[w4 * 4 + 3];

  const v4f* __restrict__ xv = (const v4f*)(x   + (size_t)nc * VOL);
  v4f* __restrict__       ov = (v4f*)      (out + (size_t)nc * VOL);

  for (int d = DIM - 1; d >= 0; --d) {
    const float ad = s_ad[d];
    #pragma unroll
    for (int j = 3; j >= 0; --j) {
      const int h = h0 + 16 * j;
      const float m = ad * s_ah[h];
      const size_t idx = (size_t)(d * DIM + h) * 16 + w4;
      v4f v = xv[idx];
      v4f r = (v * aw4) * m;
      __builtin_nontemporal_store(r, &ov[idx]);      // TH_NT b128 store
    }
  }
}

// ---------------------------------------------------------------------------
extern "C" void kernel_launch(void* const* d_in, const int* in_sizes, int n_in,
                              void* d_out, int out_size, void* d_ws, size_t ws_size,
                              hipStream_t stream) {
  (void)in_sizes; (void)n_in; (void)out_size; (void)ws_size;
  const float* x   = (const float*)d_in[0];
  const float* w1  = (const float*)d_in[1];
  const float* b1  = (const float*)d_in[2];
  const float* gma = (const float*)d_in[3];
  const float* bta = (const float*)d_in[4];
  const float* mu  = (const float*)d_in[5];
  const float* var = (const float*)d_in[6];
  const float* wd  = (const float*)d_in[7];
  const float* bd  = (const float*)d_in[8];
  const float* wh  = (const float*)d_in[9];
  const float* bh  = (const float*)d_in[10];
  const float* ww  = (const float*)d_in[11];
  const float* bw  = (const float*)d_in[12];
  float* out = (float*)d_out;

  float* pool = (float*)d_ws;                      // 4*64*192 floats = 192KB
  float* att  = pool + (size_t)NB * CB * LTOT;     // 4*64*192 floats = 192KB

  pool_kernel<<<NB * CB, 256, 0, stream>>>(x, pool);
  mid_kernel<<<NB * 12, 32, 0, stream>>>(pool, w1, b1, gma, bta, mu, var,
                                         wd, bd, wh, bh, ww, bw, att);
  apply_kernel<<<NB * CB, 256, 0, stream>>>(x, att, out);
}